// SeqSelfAttention_4140348473507
// MI455X (gfx1250) — compile-verified
//
#include <hip/hip_runtime.h>
#include <hip/hip_bf16.h>

typedef __attribute__((ext_vector_type(2))) float v2f;
typedef __attribute__((ext_vector_type(8))) float v8f;

#define L_SEQ   2048
#define F_DIM   128
#define U_DIM   32
#define HALF_W  32          // ATTENTION_WIDTH / 2
#define KSPLIT  8           // waves per query tile (key-range split)
#define KSTRIDE (16 * KSPLIT)

__device__ __forceinline__ float fast_exp2(float x) {
#if __has_builtin(__builtin_amdgcn_exp2f)
    return __builtin_amdgcn_exp2f(x);
#else
    return exp2f(x);
#endif
}
__device__ __forceinline__ float fast_tanh(float x) {
#if __has_builtin(__builtin_amdgcn_tanhf)
    return __builtin_amdgcn_tanhf(x);     // v_tanh_f32 (TRANS pipe, gfx1250)
#else
    return tanhf(x);
#endif
}
__device__ __forceinline__ float fast_rcp(float x) {
#if __has_builtin(__builtin_amdgcn_rcpf)
    return __builtin_amdgcn_rcpf(x);
#else
    return 1.0f / x;
#endif
}

// Async copy of 2KB (contiguous) global -> LDS, one b128 per lane x 4 insts.
// Instruction offset applies to BOTH the LDS and global addresses (ISA 10.x).
// Tracked by ASYNCcnt; completes in order with other async loads.
__device__ __forceinline__ void async_copy_2k(uint32_t lds_byte, const float* gsrc) {
    asm volatile(
        "global_load_async_to_lds_b128 %0, %1, off\n\t"
        "global_load_async_to_lds_b128 %0, %1, off offset:512\n\t"
        "global_load_async_to_lds_b128 %0, %1, off offset:1024\n\t"
        "global_load_async_to_lds_b128 %0, %1, off offset:1536"
        :: "v"(lds_byte), "v"(gsrc) : "memory");
}
__device__ __forceinline__ void wait_async_le4() {
    asm volatile("s_wait_asynccnt 0x4" ::: "memory");
}

// ---------------------------------------------------------------------------
// Phase 1: qv = X @ Wt + bh  (bh folded in),  kv = X @ Wx.   (2048 x 32 each)
// ---------------------------------------------------------------------------
__global__ __launch_bounds__(64) void qk_proj_kernel(
    const float* __restrict__ inputs, const float* __restrict__ Wt,
    const float* __restrict__ Wx, const float* __restrict__ bh,
    float* __restrict__ qv, float* __restrict__ kv)
{
    const int l = blockIdx.x;
    const int t = threadIdx.x;              // 64 threads: 32 for qv, 32 for kv
    const int u = t & (U_DIM - 1);
    const bool isK = t >= U_DIM;
    const float* __restrict__ W = isK ? Wx : Wt;
    const float4* __restrict__ row4 = (const float4*)(inputs + l * F_DIM);
    float acc = 0.f;
#pragma unroll
    for (int f4 = 0; f4 < F_DIM / 4; ++f4) {
        float4 x = row4[f4];
        acc = fmaf(x.x, W[(4 * f4 + 0) * U_DIM + u], acc);
        acc = fmaf(x.y, W[(4 * f4 + 1) * U_DIM + u], acc);
        acc = fmaf(x.z, W[(4 * f4 + 2) * U_DIM + u], acc);
        acc = fmaf(x.w, W[(4 * f4 + 3) * U_DIM + u], acc);
    }
    if (isK) kv[l * U_DIM + u] = acc;
    else     qv[l * U_DIM + u] = acc + bh[u];
}

// ---------------------------------------------------------------------------
// Phase 2: flash-style local attention.
// Block = 8 waves = one 16-query tile; wave w handles key chunks
// k0 = kstart + (w + j*8)*16.  Scores bounded in (0,1) => max-free softmax.
// Pipeline per chunk:
//   issue async kv copy (next chunk, alt LDS buffer)       [ASYNCcnt]
//   s_wait_asynccnt 4  -> current chunk's kv is resident
//   prefetch 64 B-operand floats into regs                 [LOADcnt, hidden]
//   256 x v_tanh score phase, 2-way interleaved chains     <- TRANS-bound
//   32 x v_wmma_f32_16x16x4_f32 back-to-back
// Operand layouts (wave32):
//   A 16x4 f32 : lane l -> M = l&15, vgpr r -> K = 2*(l>>4) + r
//   B 4x16 f32 : lane l -> N = l&15, vgpr r -> K = 2*(l>>4) + r
//   C 16x16 f32: lane l -> N = l&15, vgpr r -> M = 8*(l>>4) + r
// ---------------------------------------------------------------------------
__global__ __launch_bounds__(32 * KSPLIT) void attn_kernel(
    const float* __restrict__ inputs, const float* __restrict__ qv_g,
    const float* __restrict__ kv_g, const float* __restrict__ Wa,
    const float* __restrict__ ba, float* __restrict__ out)
{
    __shared__ __align__(16) float skv[KSPLIT][2][16 * U_DIM]; // double-buffered kv
    __shared__ float sacc[16 * F_DIM];                         // O tile accumulator
    __shared__ float srs[16];                                  // row sums

    const int tid  = threadIdx.x;
    const int lane = tid & 31;
    const int wid  = tid >> 5;
    const int q0   = blockIdx.x * 16;
    const int m    = lane & 15;             // query row (A/C role) / col n (B/C)
    const int h    = lane >> 4;
    const int n    = m;
    const int q    = q0 + m;
    const int qlo  = q - HALF_W;            // keys below this are exactly masked

    // zero the cross-wave accumulators
    for (int i = tid; i < 16 * F_DIM; i += 32 * KSPLIT) sacc[i] = 0.f;
    if (tid < 16) srs[tid] = 0.f;
    __syncthreads();

    // per-lane LDS byte addresses of the two kv buffers (low 32 bits of the
    // generic pointer are the 0-based LDS offset on gfx1250)
    const uint32_t lb0 = (uint32_t)(uintptr_t)(&skv[wid][0][0]) + (uint32_t)(lane * 16);
    const uint32_t lb1 = (uint32_t)(uintptr_t)(&skv[wid][1][0]) + (uint32_t)(lane * 16);

    // qv row for this lane's query (bh already folded in)
    float qvr[U_DIM];
    {
        const float4* src = (const float4*)(qv_g + q * U_DIM);
#pragma unroll
        for (int i = 0; i < U_DIM / 4; ++i) {
            float4 v = src[i];
            qvr[4 * i + 0] = v.x; qvr[4 * i + 1] = v.y;
            qvr[4 * i + 2] = v.z; qvr[4 * i + 3] = v.w;
        }
    }
    // Wa, ba are wave-uniform -> scalar regs / s_loads
    float wa[U_DIM];
#pragma unroll
    for (int i = 0; i < U_DIM; ++i) wa[i] = Wa[i];
    const float bias = ba[0];

    const v8f vzero = {0.f, 0.f, 0.f, 0.f, 0.f, 0.f, 0.f, 0.f};
    v8f acc[8];
#pragma unroll
    for (int f = 0; f < 8; ++f) acc[f] = vzero;
    float rowsum = 0.f;

    int kstart = q0 - HALF_W;
    if (kstart < 0) kstart = 0;

    int k0  = kstart + wid * 16;
    int buf = 0;
    if (k0 < L_SEQ)                          // prologue: stage first chunk
        async_copy_2k(lb0, kv_g + k0 * U_DIM + lane * 4);

    for (; k0 < L_SEQ; k0 += KSTRIDE) {
        // ---- issue async kv copy for the NEXT chunk into the alt buffer ----
        {
            int kn = k0 + KSTRIDE;
            if (kn >= L_SEQ) kn = k0;        // harmless dummy refetch at the tail
            async_copy_2k(buf ? lb0 : lb1, kv_g + kn * U_DIM + lane * 4);
        }
        wait_async_le4();                    // oldest 4 asyncs (this chunk) done

        // ---- prefetch all B operands for this chunk (latency hidden by the
        //      score phase below) -------------------------------------------
        v2f bp[4][8];
#pragma unroll
        for (int cc = 0; cc < 4; ++cc) {
            const float* r0 = inputs + (k0 + cc * 4 + 2 * h) * F_DIM + n;
#pragma unroll
            for (int f = 0; f < 8; ++f) {
                bp[cc][f].x = r0[f * 16];
                bp[cc][f].y = r0[F_DIM + f * 16];
            }
        }

        // ---- 8 score elements per lane in WMMA A layout, computed as 4
        //      pairs of interleaved chains so every TRANS slot has an
        //      independent consumer (kills the v_nop bubbles) ----------------
        const float* kvbase = &skv[wid][buf][0];
        float p8[8];
#pragma unroll
        for (int pp = 0; pp < 4; ++pp) {
            const int kl0 = pp * 4 + 2 * h;              // rows for (p, p+1)
            const float4* kr0 = (const float4*)(kvbase + (kl0 + 0) * U_DIM);
            const float4* kr1 = (const float4*)(kvbase + (kl0 + 1) * U_DIM);
            float e0 = bias, e1 = bias;
#pragma unroll
            for (int i = 0; i < U_DIM / 4; ++i) {
                float4 c0 = kr0[i];
                float4 c1 = kr1[i];
                float t0, t1;
                t0 = fast_tanh(qvr[4*i+0] + c0.x);
                t1 = fast_tanh(qvr[4*i+0] + c1.x);
                e0 = fmaf(wa[4*i+0], t0, e0);
                e1 = fmaf(wa[4*i+0], t1, e1);
                t0 = fast_tanh(qvr[4*i+1] + c0.y);
                t1 = fast_tanh(qvr[4*i+1] + c1.y);
                e0 = fmaf(wa[4*i+1], t0, e0);
                e1 = fmaf(wa[4*i+1], t1, e1);
                t0 = fast_tanh(qvr[4*i+2] + c0.z);
                t1 = fast_tanh(qvr[4*i+2] + c1.z);
                e0 = fmaf(wa[4*i+2], t0, e0);
                e1 = fmaf(wa[4*i+2], t1, e1);
                t0 = fast_tanh(qvr[4*i+3] + c0.w);
                t1 = fast_tanh(qvr[4*i+3] + c1.w);
                e0 = fmaf(wa[4*i+3], t0, e0);
                e1 = fmaf(wa[4*i+3], t1, e1);
            }
            // sigmoid in (0,1) -> p = exp(sigmoid) in (1,e): no max tracking
            float s0 = fast_rcp(1.0f + fast_exp2(-1.44269504f * e0));
            float s1 = fast_rcp(1.0f + fast_exp2(-1.44269504f * e1));
            float pv0 = fast_exp2(1.44269504f * s0);
            float pv1 = fast_exp2(1.44269504f * s1);
            if (k0 + kl0 + 0 < qlo) pv0 = 0.f;   // left-side-only penalty
            if (k0 + kl0 + 1 < qlo) pv1 = 0.f;
            p8[2 * pp + 0] = pv0;
            p8[2 * pp + 1] = pv1;
        }

        // ---- row sum (lane ^ 16 holds the other 8 keys of this row) --------
        float ls = 0.f;
#pragma unroll
        for (int p = 0; p < 8; ++p) ls += p8[p];
        ls += __shfl_xor(ls, 16, 32);
        rowsum += ls;

        // ---- O += P @ V : 32 x WMMA f32 16x16x4, operands all in regs ------
        v2f a[4];
#pragma unroll
        for (int cc = 0; cc < 4; ++cc) { a[cc].x = p8[2 * cc]; a[cc].y = p8[2 * cc + 1]; }
#pragma unroll
        for (int cc = 0; cc < 4; ++cc)
#pragma unroll
            for (int f = 0; f < 8; ++f)
                acc[f] = __builtin_amdgcn_wmma_f32_16x16x4_f32(
                    false, a[cc], false, bp[cc][f], (short)0, acc[f], false, false);

        buf ^= 1;
    }

    // ---- combine partials across the 8 waves (LDS float atomics) ----------
#pragma unroll
    for (int f = 0; f < 8; ++f)
#pragma unroll
        for (int r = 0; r < 8; ++r)
            atomicAdd(&sacc[(8 * h + r) * F_DIM + f * 16 + n], acc[f][r]);
    if (h == 0) atomicAdd(&srs[m], rowsum);
    __syncthreads();

    // ---- normalize + store 16x128 output tile ------------------------------
    for (int i = tid; i < 16 * F_DIM; i += 32 * KSPLIT) {
        const int row = i >> 7;             // /F_DIM
        const int col = i & (F_DIM - 1);
        out[(q0 + row) * F_DIM + col] = sacc[i] * fast_rcp(srs[row]);
    }
}

// ---------------------------------------------------------------------------
extern "C" void kernel_launch(void* const* d_in, const int* in_sizes, int n_in,
                              void* d_out, int out_size, void* d_ws, size_t ws_size,
                              hipStream_t stream)
{
    const float* inputs = (const float*)d_in[0];
    const float* Wt     = (const float*)d_in[1];
    const float* Wx     = (const float*)d_in[2];
    const float* bh     = (const float*)d_in[3];
    const float* Wa     = (const float*)d_in[4];
    const float* ba     = (const float*)d_in[5];
    float* out = (float*)d_out;

    float* qv = (float*)d_ws;                       // 2048*32 f32
    float* kv = qv + L_SEQ * U_DIM;                 // 2048*32 f32

    qk_proj_kernel<<<L_SEQ, 64, 0, stream>>>(inputs, Wt, Wx, bh, qv, kv);
    attn_kernel<<<L_SEQ / 16, 32 * KSPLIT, 0, stream>>>(inputs, qv, kv, Wa, ba, out);
}